// TransformerBlock_39419209842958
// MI455X (gfx1250) — compile-verified
//
#include <hip/hip_runtime.h>
#include <math.h>

// Problem constants (from reference): B=2, S=2048, H=768, NH=12, HD=64, FFN=3072
#define B_   2
#define S_   2048
#define H_   768
#define NH_  12
#define HD_  64
#define FFN_ 3072
#define M_   (B_ * S_)        // 4096 rows in every GEMM
#define QKVLD (3 * H_)        // 2304

typedef __attribute__((ext_vector_type(2))) float v2f;
typedef __attribute__((ext_vector_type(4))) float v4f;
typedef __attribute__((ext_vector_type(8))) float v8f;

// ---------------------------------------------------------------------------
// Full-precision fp32 WMMA: D(16x16) = A(16x4) * B(4x16) + C, wave32.
// A frag: lane<16 -> (M=lane, K=kk..kk+1), lane>=16 -> (M=lane-16, K=kk+2..kk+3)
// B frag: lane<16 -> (K=kk, N=lane) in v0 / (K=kk+1) in v1; hi lanes K=kk+2/kk+3
// C/D  : VGPR r, lane<16 -> (M=r, N=lane); lane>=16 -> (M=r+8, N=lane-16)
// ---------------------------------------------------------------------------
__device__ __forceinline__ v8f wmma4(v2f a, v2f b, v8f c) {
  return __builtin_amdgcn_wmma_f32_16x16x4_f32(false, a, false, b, (short)0, c,
                                               false, false);
}

__device__ __forceinline__ float gelu_exact(float x) {
  return 0.5f * x * (1.0f + erff(x * 0.70710678118654752440f));
}

// ---------------------------------------------------------------------------
// LayerNorm: one block (256 threads) per row of H=768.
// ---------------------------------------------------------------------------
__global__ __launch_bounds__(256) void ln_kernel(const float* __restrict__ x,
                                                 const float* __restrict__ g,
                                                 const float* __restrict__ b,
                                                 float* __restrict__ o) {
  __shared__ float red[256];
  const int t = threadIdx.x;
  const size_t base = (size_t)blockIdx.x * H_;

  float s = 0.0f;
  for (int i = t; i < H_; i += 256) s += x[base + i];
  red[t] = s;
  __syncthreads();
  for (int off = 128; off > 0; off >>= 1) {
    if (t < off) red[t] += red[t + off];
    __syncthreads();
  }
  const float mu = red[0] * (1.0f / (float)H_);
  __syncthreads();

  float vs = 0.0f;
  for (int i = t; i < H_; i += 256) {
    float d = x[base + i] - mu;
    vs += d * d;
  }
  red[t] = vs;
  __syncthreads();
  for (int off = 128; off > 0; off >>= 1) {
    if (t < off) red[t] += red[t + off];
    __syncthreads();
  }
  const float rstd = rsqrtf(red[0] * (1.0f / (float)H_) + 1e-5f);

  for (int i = t; i < H_; i += 256)
    o[base + i] = (x[base + i] - mu) * rstd * g[i] + b[i];
}

// ---------------------------------------------------------------------------
// GEMM: C[M,N] = A[M,K] * W[N,K]^T + bias[N]  (+ resid)  (+ exact GELU)
// Block tile 64x64, BK=16, double-buffered LDS staging (one barrier/stage,
// global loads for stage s+1 in flight while stage s computes on WMMA).
// Padded LDS stride 20 -> conflict-free b64 fragment reads. 8 waves; each wave
// owns a 16(M) x 32(N) piece as two f32 16x16 WMMA accumulators.
// ---------------------------------------------------------------------------
#define GLDA 20

__global__ __launch_bounds__(256) void gemm_kernel(
    const float* __restrict__ A, const float* __restrict__ W,
    const float* __restrict__ bias, const float* __restrict__ resid,
    float* __restrict__ C, int M, int N, int K, int act) {
  __shared__ float As[2][64 * GLDA];
  __shared__ float Ws[2][64 * GLDA];

  const int t = threadIdx.x;
  const int wave = t >> 5, lane = t & 31;
  const int m0 = blockIdx.x * 64, n0 = blockIdx.y * 64;
  const int wm = 16 * (wave & 3);        // 0,16,32,48
  const int wn = 32 * (wave >> 2);       // 0,32
  const int lm = lane & 15;
  const int koff = (lane < 16) ? 0 : 2;
  const int rbase = (lane < 16) ? 0 : 8;

  const int lrow = t >> 2;               // 0..63
  const int lcol = (t & 3) * 4;          // 0,4,8,12

  const float* Ap = A + (size_t)(m0 + lrow) * K + lcol;
  const float* Wp = W + (size_t)(n0 + lrow) * K + lcol;
  const int lds_off = lrow * GLDA + lcol;

  v8f acc0 = {};
  v8f acc1 = {};

  // Prologue: stage 0 into buffer 0.
  {
    v4f av = *(const v4f*)(Ap);
    v4f wv = *(const v4f*)(Wp);
    *(v4f*)(As[0] + lds_off) = av;
    *(v4f*)(Ws[0] + lds_off) = wv;
  }
  __syncthreads();

  const int nstages = K >> 4;
  for (int s = 0; s < nstages; ++s) {
    const int p = s & 1;
    const bool has_next = (s + 1) < nstages;

    // Issue next stage's global loads now; they stay in flight during WMMA.
    v4f av, wv;
    if (has_next) {
      av = *(const v4f*)(Ap + (size_t)(s + 1) * 16);
      wv = *(const v4f*)(Wp + (size_t)(s + 1) * 16);
    }

    const float* Ab = As[p];
    const float* Wb = Ws[p];
#pragma unroll
    for (int kk = 0; kk < 16; kk += 4) {
      v2f a  = *(const v2f*)(Ab + (wm + lm) * GLDA + kk + koff);
      v2f b0 = *(const v2f*)(Wb + (wn + lm) * GLDA + kk + koff);
      v2f b1 = *(const v2f*)(Wb + (wn + 16 + lm) * GLDA + kk + koff);
      acc0 = wmma4(a, b0, acc0);
      acc1 = wmma4(a, b1, acc1);
    }

    // Writers of buffer 1-p never race readers of buffer p -> one barrier.
    if (has_next) {
      *(v4f*)(As[1 - p] + lds_off) = av;
      *(v4f*)(Ws[1 - p] + lds_off) = wv;
    }
    __syncthreads();
  }

#pragma unroll
  for (int r = 0; r < 8; ++r) {
    const int m = m0 + wm + r + rbase;
    const int n = n0 + wn + lm;
    const size_t i0 = (size_t)m * N + n;
    float v0 = acc0[r] + bias[n];
    float v1 = acc1[r] + bias[n + 16];
    if (resid) {
      v0 += resid[i0];
      v1 += resid[i0 + 16];
    }
    if (act == 1) {
      v0 = gelu_exact(v0);
      v1 = gelu_exact(v1);
    }
    C[i0] = v0;
    C[i0 + 16] = v1;
  }
}

// ---------------------------------------------------------------------------
// Flash attention (fp32 WMMA everywhere). One block = 64 query rows of one
// (batch, head). Causal mask applied analytically; online softmax in fp32.
// qkv layout per row: [q(768) | k(768) | v(768)], head slice nh*64.
// ---------------------------------------------------------------------------
#define ALD 66

__global__ __launch_bounds__(256) void attn_kernel(
    const float* __restrict__ qkv, float* __restrict__ out) {
  __shared__ float Qs[64 * ALD];
  __shared__ float Ks[64 * ALD];   // reused as P (probabilities) each kv tile
  __shared__ float Vs[64 * ALD];
  __shared__ float alpha_s[64];
  __shared__ float lrow_s[64];

  const int t = threadIdx.x;
  const int wave = t >> 5, lane = t & 31;
  const int lm = lane & 15;
  const int koff = (lane < 16) ? 0 : 2;
  const int rbase = (lane < 16) ? 0 : 8;
  const int wm = 16 * (wave & 3);
  const int wn = 32 * (wave >> 2);

  const int qt = blockIdx.x;                 // 0..31 query tile
  const int bh = blockIdx.y;                 // 0..23
  const int b = bh / NH_, nh = bh % NH_;
  const int q0 = qt * 64;

  const float* qbase = qkv + (size_t)b * S_ * QKVLD + nh * HD_;
  const float* kbase = qbase + H_;
  const float* vbase = qbase + 2 * H_;
  float* Ps = Ks;

  // Load Q tile (64 rows x 64 cols) as float2s.
  for (int i = 0; i < 8; ++i) {
    int f = t + 256 * i;                     // float2 index 0..2047
    int r = f >> 5;
    int c2 = (f & 31) * 2;
    *(v2f*)(Qs + r * ALD + c2) =
        *(const v2f*)(qbase + (size_t)(q0 + r) * QKVLD + c2);
  }

  v8f o0 = {};
  v8f o1 = {};
  float mi = -INFINITY, li = 0.0f;           // valid for t < 64

  for (int j = 0; j <= qt; ++j) {
    const int k0 = j * 64;
    for (int i = 0; i < 8; ++i) {
      int f = t + 256 * i;
      int r = f >> 5;
      int c2 = (f & 31) * 2;
      *(v2f*)(Ks + r * ALD + c2) =
          *(const v2f*)(kbase + (size_t)(k0 + r) * QKVLD + c2);
      *(v2f*)(Vs + r * ALD + c2) =
          *(const v2f*)(vbase + (size_t)(k0 + r) * QKVLD + c2);
    }
    __syncthreads();

    // S = Q * K^T  (scaled by 1/sqrt(64) at store)
    v8f s0 = {};
    v8f s1 = {};
#pragma unroll
    for (int kk = 0; kk < 64; kk += 4) {
      v2f a  = *(const v2f*)(Qs + (wm + lm) * ALD + kk + koff);
      v2f b0 = *(const v2f*)(Ks + (wn + lm) * ALD + kk + koff);
      v2f b1 = *(const v2f*)(Ks + (wn + 16 + lm) * ALD + kk + koff);
      s0 = wmma4(a, b0, s0);
      s1 = wmma4(a, b1, s1);
    }
    __syncthreads();                          // everyone done reading Ks

#pragma unroll
    for (int r = 0; r < 8; ++r) {
      Ps[(wm + r + rbase) * ALD + wn + lm]      = s0[r] * 0.125f;
      Ps[(wm + r + rbase) * ALD + wn + 16 + lm] = s1[r] * 0.125f;
    }
    __syncthreads();

    // Online softmax: one thread per query row.
    if (t < 64) {
      const int m = t;
      float* Pr = Ps + m * ALD;
      const int limit = (j == qt) ? (m + 1) : 64;   // causal
      float mx = -INFINITY;
      for (int c = 0; c < limit; ++c) mx = fmaxf(mx, Pr[c]);
      const float mnew = fmaxf(mi, mx);
      const float al = expf(mi - mnew);             // 0 when mi == -inf
      float sum = 0.0f;
      for (int c = 0; c < 64; ++c) {
        float p = (c < limit) ? expf(Pr[c] - mnew) : 0.0f;
        Pr[c] = p;
        sum += p;
      }
      li = li * al + sum;
      mi = mnew;
      alpha_s[m] = al;
    }
    __syncthreads();

    // Rescale running O, then O += P * V.
#pragma unroll
    for (int r = 0; r < 8; ++r) {
      const float al = alpha_s[wm + r + rbase];
      o0[r] *= al;
      o1[r] *= al;
    }
#pragma unroll
    for (int kk = 0; kk < 64; kk += 4) {
      v2f a = *(const v2f*)(Ps + (wm + lm) * ALD + kk + koff);
      const int kr = kk + koff;
      v2f b0, b1;
      b0.x = Vs[kr * ALD + wn + lm];
      b0.y = Vs[(kr + 1) * ALD + wn + lm];
      b1.x = Vs[kr * ALD + wn + 16 + lm];
      b1.y = Vs[(kr + 1) * ALD + wn + 16 + lm];
      o0 = wmma4(a, b0, o0);
      o1 = wmma4(a, b1, o1);
    }
    __syncthreads();                          // before Ks/Vs are reloaded
  }

  if (t < 64) lrow_s[t] = li;
  __syncthreads();

#pragma unroll
  for (int r = 0; r < 8; ++r) {
    const int row = wm + r + rbase;
    const float inv = 1.0f / lrow_s[row];
    const size_t o = ((size_t)b * S_ + q0 + row) * H_ + nh * HD_ + wn + lm;
    out[o] = o0[r] * inv;
    out[o + 16] = o1[r] * inv;
  }
}

// Trailing tuple scalar (jnp.float32(0.0)).
__global__ void tail_kernel(float* out, int idx) {
  if (threadIdx.x == 0 && blockIdx.x == 0) out[idx] = 0.0f;
}

// ---------------------------------------------------------------------------
extern "C" void kernel_launch(void* const* d_in, const int* in_sizes, int n_in,
                              void* d_out, int out_size, void* d_ws,
                              size_t ws_size, hipStream_t stream) {
  (void)in_sizes; (void)n_in; (void)ws_size;
  const float* x     = (const float*)d_in[0];
  // d_in[1] = causal mask (int) — applied analytically, not needed.
  const float* ln1_g = (const float*)d_in[2];
  const float* ln1_b = (const float*)d_in[3];
  const float* qkv_w = (const float*)d_in[4];
  const float* qkv_b = (const float*)d_in[5];
  const float* out_w = (const float*)d_in[6];
  const float* out_b = (const float*)d_in[7];
  const float* ln2_g = (const float*)d_in[8];
  const float* ln2_b = (const float*)d_in[9];
  const float* fc1_w = (const float*)d_in[10];
  const float* fc1_b = (const float*)d_in[11];
  const float* fc2_w = (const float*)d_in[12];
  const float* fc2_b = (const float*)d_in[13];

  float* xo = (float*)d_out;                  // also holds x after residual 1
  float* ws = (float*)d_ws;
  float* h    = ws;                           //  M*H      (12 MB)
  float* qkvb = h + (size_t)M_ * H_;          //  M*3H     (36 MB)
  float* attn = qkvb + (size_t)M_ * 3 * H_;   //  M*H      (12 MB)
  float* ffn  = attn + (size_t)M_ * H_;       //  M*FFN    (48 MB)

  // h = LN1(x)
  ln_kernel<<<M_, 256, 0, stream>>>(x, ln1_g, ln1_b, h);
  // qkv = h @ qkv_w^T + qkv_b
  gemm_kernel<<<dim3(M_ / 64, (3 * H_) / 64), 256, 0, stream>>>(
      h, qkv_w, qkv_b, nullptr, qkvb, M_, 3 * H_, H_, 0);
  // attn = causal softmax(q k^T / 8) v
  attn_kernel<<<dim3(S_ / 64, B_ * NH_), 256, 0, stream>>>(qkvb, attn);
  // xo = x + attn @ out_w^T + out_b
  gemm_kernel<<<dim3(M_ / 64, H_ / 64), 256, 0, stream>>>(
      attn, out_w, out_b, x, xo, M_, H_, H_, 0);
  // h = LN2(xo)
  ln_kernel<<<M_, 256, 0, stream>>>(xo, ln2_g, ln2_b, h);
  // ffn = gelu(h @ fc1_w^T + fc1_b)
  gemm_kernel<<<dim3(M_ / 64, FFN_ / 64), 256, 0, stream>>>(
      h, fc1_w, fc1_b, nullptr, ffn, M_, FFN_, H_, 1);
  // xo = xo + ffn @ fc2_w^T + fc2_b   (in-place residual: 1 read + 1 write per elem)
  gemm_kernel<<<dim3(M_ / 64, H_ / 64), 256, 0, stream>>>(
      ffn, fc2_w, fc2_b, xo, xo, M_, H_, FFN_, 0);
  // tuple's trailing 0.0f scalar
  tail_kernel<<<1, 32, 0, stream>>>(xo, out_size - 1);
}